// GCN_57921928954524
// MI455X (gfx1250) — compile-verified
//
#include <hip/hip_runtime.h>
#include <hip/hip_bf16.h>

#define N_NODES   100000
#define N_EDGES   3200000
#define NUM_GRAPHS 1024
#define F_IN      128
#define EMB       32
#define N_HIDDEN  3

typedef __attribute__((ext_vector_type(2))) float v2f;
typedef __attribute__((ext_vector_type(8))) float v8f;

// ---------------------------------------------------------------------------
// WMMA fp32 GEMM: out[N,32] = A[N,K] @ W[K,32]
// One wave computes a 16-row x 32-col tile via two 16x16 accumulators,
// stepping K by 4 with V_WMMA_F32_16X16X4_F32 (full fp32 matrix path).
// ---------------------------------------------------------------------------
__global__ void gcn_gemm_wmma(const float* __restrict__ A,
                              const float* __restrict__ W,
                              float* __restrict__ out,
                              int nRows, int K) {
  const int lane = threadIdx.x & 31;
  const int wave = threadIdx.x >> 5;
  const int wavesPerBlock = blockDim.x >> 5;
  const int tile = blockIdx.x * wavesPerBlock + wave;   // 16-row tile id
  const int row0 = tile * 16;
  if (row0 >= nRows) return;                            // wave-uniform: EXEC stays full

  const int m     = lane & 15;          // row (A) / col (B) within tile
  const int kHalf = (lane >> 4) << 1;   // lanes 0-15 -> K{0,1}; lanes 16-31 -> K{2,3}

  v8f c0 = {};  // cols 0..15
  v8f c1 = {};  // cols 16..31

  const float* __restrict__ arow = A + (size_t)(row0 + m) * (size_t)K;

  for (int k = 0; k < K; k += 4) {
    v2f a;
    a.x = arow[k + kHalf];
    a.y = arow[k + kHalf + 1];

    const float* w0 = W + (size_t)(k + kHalf) * 32;
    const float* w1 = W + (size_t)(k + kHalf + 1) * 32;
    v2f b0, b1;
    b0.x = w0[m];       b0.y = w1[m];
    b1.x = w0[m + 16];  b1.y = w1[m + 16];

    c0 = __builtin_amdgcn_wmma_f32_16x16x4_f32(false, a, false, b0,
                                               (short)0, c0, false, false);
    c1 = __builtin_amdgcn_wmma_f32_16x16x4_f32(false, a, false, b1,
                                               (short)0, c1, false, false);
  }

  // D layout: VGPR r -> M = r (lanes 0-15), M = r+8 (lanes 16-31); N = lane&15
  const int rowBase = row0 + ((lane >> 4) << 3);
  const int col = lane & 15;
#pragma unroll
  for (int r = 0; r < 8; ++r) {
    const int row = rowBase + r;
    if (row < nRows) {
      out[(size_t)row * 32 + col]      = c0[r];
      out[(size_t)row * 32 + col + 16] = c1[r];
    }
  }
}

// ---------------------------------------------------------------------------
// Degree / normalization helpers
// ---------------------------------------------------------------------------
__global__ void fill_f32(float* p, float v, int n) {
  int i = blockIdx.x * blockDim.x + threadIdx.x;
  if (i < n) p[i] = v;
}

__global__ void deg_accum(const int* __restrict__ dst, float* __restrict__ deg, int nE) {
  int e = blockIdx.x * blockDim.x + threadIdx.x;
  if (e < nE) atomicAdd(&deg[dst[e]], 1.0f);
}

__global__ void rsqrt_inplace(float* p, int n) {
  int i = blockIdx.x * blockDim.x + threadIdx.x;
  if (i < n) p[i] = rsqrtf(fmaxf(p[i], 1.0f));
}

// agg = hW * dinv^2  (self-loop contribution: norm(i,i) = dinv[i]*dinv[i])
__global__ void init_agg(const float* __restrict__ hW, const float* __restrict__ dinv,
                         float* __restrict__ agg, int nNodes) {
  int idx = blockIdx.x * blockDim.x + threadIdx.x;
  if (idx < nNodes * 32) {
    int i = idx >> 5;
    float d = dinv[i];
    agg[idx] = hW[idx] * d * d;
  }
}

// 8 threads per edge, float4 per thread: agg[dst] += hW[src] * dinv[src]*dinv[dst]
__global__ void edge_scatter(const int* __restrict__ src, const int* __restrict__ dst,
                             const float* __restrict__ dinv, const float* __restrict__ hW,
                             float* __restrict__ agg, int nE) {
  int t = blockIdx.x * blockDim.x + threadIdx.x;
  int e = t >> 3;
  if (e >= nE) return;
  int f4 = (t & 7) << 2;
  int s = src[e];
  int d = dst[e];
  float nrm = dinv[s] * dinv[d];
  const float4 v = *(const float4*)(hW + (size_t)s * 32 + f4);
  float* o = agg + (size_t)d * 32 + f4;
  atomicAdd(o + 0, v.x * nrm);
  atomicAdd(o + 1, v.y * nrm);
  atomicAdd(o + 2, v.z * nrm);
  atomicAdd(o + 3, v.w * nrm);
}

__global__ void bias_act(const float* __restrict__ agg, const float* __restrict__ b,
                         float* __restrict__ h, int nNodes) {
  int idx = blockIdx.x * blockDim.x + threadIdx.x;
  if (idx < nNodes * 32) {
    float v = agg[idx] + b[idx & 31];
    h[idx] = (v >= 0.0f) ? v : 0.01f * v;
  }
}

// ---------------------------------------------------------------------------
// Global max pool via order-preserving uint encoding + atomicMax
// ---------------------------------------------------------------------------
__device__ __forceinline__ unsigned enc_f(float f) {
  unsigned u = __float_as_uint(f);
  return (u & 0x80000000u) ? ~u : (u | 0x80000000u);
}
__device__ __forceinline__ float dec_f(unsigned e) {
  unsigned u = (e & 0x80000000u) ? (e & 0x7FFFFFFFu) : ~e;
  return __uint_as_float(u);
}

__global__ void fill_u32(unsigned* p, unsigned v, int n) {
  int i = blockIdx.x * blockDim.x + threadIdx.x;
  if (i < n) p[i] = v;
}

__global__ void pool_max(const float* __restrict__ h, const int* __restrict__ batch,
                         unsigned* __restrict__ pe, int nNodes) {
  int idx = blockIdx.x * blockDim.x + threadIdx.x;
  if (idx < nNodes * 32) {
    int i = idx >> 5;
    int f = idx & 31;
    atomicMax(&pe[(size_t)batch[i] * 32 + f], enc_f(h[idx]));
  }
}

__global__ void decode_pool(const unsigned* __restrict__ pe, float* __restrict__ pooled, int n) {
  int i = blockIdx.x * blockDim.x + threadIdx.x;
  if (i < n) pooled[i] = dec_f(pe[i]);
}

__global__ void final_out(const float* __restrict__ pooled, const float* __restrict__ Wout,
                          const float* __restrict__ bout, float* __restrict__ out, int nG) {
  int g = blockIdx.x * blockDim.x + threadIdx.x;
  if (g < nG) {
    float s = 0.0f;
#pragma unroll
    for (int f = 0; f < 32; ++f) s += pooled[(size_t)g * 32 + f] * Wout[f];
    out[g] = s + bout[0];
  }
}

// ---------------------------------------------------------------------------
extern "C" void kernel_launch(void* const* d_in, const int* in_sizes, int n_in,
                              void* d_out, int out_size, void* d_ws, size_t ws_size,
                              hipStream_t stream) {
  const float* x     = (const float*)d_in[0];
  const int*   edge  = (const int*)d_in[1];
  const int*   src   = edge;
  const int*   dst   = edge + N_EDGES;
  const int*   batch = (const int*)d_in[2];
  const float* W0    = (const float*)d_in[3];
  const float* b0    = (const float*)d_in[4];
  const float* Ws    = (const float*)d_in[5];
  const float* bs    = (const float*)d_in[6];
  const float* Wout  = (const float*)d_in[7];
  const float* bout  = (const float*)d_in[8];

  float* outp = (float*)d_out;                 // [0:1024) = out, [1024:1024+32768) = pooled

  // workspace layout
  float*    dinv = (float*)d_ws;               // 100352 (padded) floats; holds deg then dinv
  float*    hW   = dinv + 100352;              // [N_NODES*32]
  float*    h    = hW   + (size_t)N_NODES * 32;
  float*    agg  = h    + (size_t)N_NODES * 32;
  unsigned* pe   = (unsigned*)(agg + (size_t)N_NODES * 32);  // [1024*32]

  const int TPB = 256;
  const int nodeElems  = N_NODES * 32;
  const int nodeBlocks = (nodeElems + TPB - 1) / TPB;
  const int edgeThr    = N_EDGES * 8;
  const int edgeBlocks = (edgeThr + TPB - 1) / TPB;

  // --- degrees & symmetric normalization ---
  fill_f32<<<(N_NODES + TPB - 1) / TPB, TPB, 0, stream>>>(dinv, 1.0f, N_NODES); // self loop
  deg_accum<<<(N_EDGES + TPB - 1) / TPB, TPB, 0, stream>>>(dst, dinv, N_EDGES);
  rsqrt_inplace<<<(N_NODES + TPB - 1) / TPB, TPB, 0, stream>>>(dinv, N_NODES);

  // --- GEMM grid: 4 waves per 128-thread block, one 16-row tile per wave ---
  const int tiles      = (N_NODES + 15) / 16;        // 6250
  const int gemmBlocks = (tiles + 3) / 4;            // 1563

  // --- layer 0 (K = 128) ---
  gcn_gemm_wmma<<<gemmBlocks, 128, 0, stream>>>(x, W0, hW, N_NODES, F_IN);
  init_agg<<<nodeBlocks, TPB, 0, stream>>>(hW, dinv, agg, N_NODES);
  edge_scatter<<<edgeBlocks, TPB, 0, stream>>>(src, dst, dinv, hW, agg, N_EDGES);
  bias_act<<<nodeBlocks, TPB, 0, stream>>>(agg, b0, h, N_NODES);

  // --- hidden layers (K = 32) ---
  for (int l = 0; l < N_HIDDEN; ++l) {
    gcn_gemm_wmma<<<gemmBlocks, 128, 0, stream>>>(h, Ws + (size_t)l * EMB * EMB,
                                                  hW, N_NODES, EMB);
    init_agg<<<nodeBlocks, TPB, 0, stream>>>(hW, dinv, agg, N_NODES);
    edge_scatter<<<edgeBlocks, TPB, 0, stream>>>(src, dst, dinv, hW, agg, N_EDGES);
    bias_act<<<nodeBlocks, TPB, 0, stream>>>(agg, bs + (size_t)l * EMB, h, N_NODES);
  }

  // --- global max pool + readout ---
  const int poolN = NUM_GRAPHS * 32;
  fill_u32<<<(poolN + TPB - 1) / TPB, TPB, 0, stream>>>(pe, 0x007FFFFFu, poolN); // enc(-inf)
  pool_max<<<nodeBlocks, TPB, 0, stream>>>(h, batch, pe, N_NODES);
  decode_pool<<<(poolN + TPB - 1) / TPB, TPB, 0, stream>>>(pe, outp + NUM_GRAPHS, poolN);
  final_out<<<(NUM_GRAPHS + TPB - 1) / TPB, TPB, 0, stream>>>(outp + NUM_GRAPHS, Wout, bout,
                                                              outp, NUM_GRAPHS);
}